// HeteroGCNLayer_10496900072194
// MI455X (gfx1250) — compile-verified
//
#include <hip/hip_runtime.h>

typedef __attribute__((ext_vector_type(2))) float v2f;
typedef __attribute__((ext_vector_type(8))) float v8f;

#define DFEAT 128

// ---------------------------------------------------------------------------
// GEMM: out[row, 0:128] = H[row, 0:128] @ W[128,128] + bias[128]
// One wave -> 16 rows x 128 cols, via V_WMMA_F32_16X16X4_F32.
//   A 16x4 f32 layout : lanes 0-15 (M=lane)  hold K=k0,k0+1 in v[0],v[1]
//                       lanes 16-31 (M=lane-16) hold K=k0+2,k0+3
//   B 4x16 f32 layout : VGPR0 = {K=k0 | K=k0+2}, VGPR1 = {K=k0+1 | K=k0+3},
//                       N = lane%16  (row-striped across lanes, half split on K)
//   C/D 16x16 f32     : VGPR r -> (M=r, N=lane) lanes 0-15 ; (M=r+8) lanes 16-31
// ---------------------------------------------------------------------------
__global__ __launch_bounds__(128) void gemm128_bias_wmma(
    const float* __restrict__ H, const float* __restrict__ W,
    const float* __restrict__ bias, float* __restrict__ out, int n_rows)
{
    const int lane = threadIdx.x & 31;
    const int wave = threadIdx.x >> 5;                 // 4 waves / block
    const int row0 = (blockIdx.x * 4 + wave) * 16;     // 16-row strip per wave
    if (row0 >= n_rows) return;                        // wave-uniform exit
    const int half = lane >> 4;                        // 0 | 1
    const int l16  = lane & 15;

    const float* __restrict__ arow = H + (size_t)(row0 + l16) * DFEAT;

    v8f acc[8];
#pragma unroll
    for (int j = 0; j < 8; ++j) acc[j] = (v8f)(0.0f);

    for (int k0 = 0; k0 < DFEAT; k0 += 4) {
        const int kk = k0 + 2 * half;
        v2f a;
        a.x = arow[kk];
        a.y = arow[kk + 1];

        v2f bf[8];
#pragma unroll
        for (int j = 0; j < 8; ++j) {
            bf[j].x = W[(size_t)kk * DFEAT + j * 16 + l16];
            bf[j].y = W[(size_t)(kk + 1) * DFEAT + j * 16 + l16];
        }
#pragma unroll
        for (int j = 0; j < 8; ++j) {
            acc[j] = __builtin_amdgcn_wmma_f32_16x16x4_f32(
                false, a, false, bf[j], (short)0, acc[j], false, false);
        }
    }

#pragma unroll
    for (int j = 0; j < 8; ++j) {
        const float bv = bias[j * 16 + l16];
#pragma unroll
        for (int r = 0; r < 8; ++r) {
            out[(size_t)(row0 + r + 8 * half) * DFEAT + j * 16 + l16] =
                acc[j][r] + bv;
        }
    }
}

// ---------------------------------------------------------------------------
// Scatter-add: sum[dst[e], f] += msg_src[src[e], f]; deg[dst[e]] += 1
// 1 thread per (edge, feature).
// ---------------------------------------------------------------------------
__global__ __launch_bounds__(256) void scatter_accum(
    const float* __restrict__ msg_src, const int* __restrict__ src_idx,
    const int* __restrict__ dst_idx, float* __restrict__ sum,
    float* __restrict__ deg, long long nE)
{
    long long gid = (long long)blockIdx.x * blockDim.x + threadIdx.x;
    if (gid >= nE * DFEAT) return;
    long long e = gid >> 7;
    int f = (int)(gid & (DFEAT - 1));
    int s = src_idx[e];
    int d = dst_idx[e];
    atomicAdd(&sum[(size_t)d * DFEAT + f], msg_src[(size_t)s * DFEAT + f]);
    if (f == 0) atomicAdd(&deg[d], 1.0f);
}

// ---------------------------------------------------------------------------
// Combine (op nodes): out = relu(proj + sum1/max(deg1,1) + sum2/max(deg2,1))
// proj lives in `out` already (in-place).
// ---------------------------------------------------------------------------
__global__ __launch_bounds__(256) void combine2_relu(
    float* __restrict__ out, const float* __restrict__ sum1,
    const float* __restrict__ deg1, const float* __restrict__ sum2,
    const float* __restrict__ deg2, long long n_elems)
{
    long long gid = (long long)blockIdx.x * blockDim.x + threadIdx.x;
    if (gid >= n_elems) return;
    long long node = gid >> 7;
    float d1 = fmaxf(deg1[node], 1.0f);
    float d2 = fmaxf(deg2[node], 1.0f);
    float v  = out[gid] + sum1[gid] / d1 + sum2[gid] / d2;
    out[gid] = fmaxf(v, 0.0f);
}

__global__ __launch_bounds__(256) void combine1_relu(
    float* __restrict__ out, const float* __restrict__ sum1,
    const float* __restrict__ deg1, long long n_elems)
{
    long long gid = (long long)blockIdx.x * blockDim.x + threadIdx.x;
    if (gid >= n_elems) return;
    long long node = gid >> 7;
    float d1 = fmaxf(deg1[node], 1.0f);
    float v  = out[gid] + sum1[gid] / d1;
    out[gid] = fmaxf(v, 0.0f);
}

// ---------------------------------------------------------------------------
extern "C" void kernel_launch(void* const* d_in, const int* in_sizes, int n_in,
                              void* d_out, int out_size, void* d_ws, size_t ws_size,
                              hipStream_t stream)
{
    const float* H_op = (const float*)d_in[0];
    const float* H_m  = (const float*)d_in[1];
    const int*   E_seq  = (const int*)d_in[2];   // [2, E] : row0=src, row1=dst
    const int*   E_op2m = (const int*)d_in[3];   // [2, E] : row0=src_op, row1=dst_m
    const float* W_op = (const float*)d_in[4];
    const float* b_op = (const float*)d_in[5];
    const float* W_m  = (const float*)d_in[6];
    const float* b_m  = (const float*)d_in[7];

    const int n_op = in_sizes[0] / DFEAT;
    const int n_m  = in_sizes[1] / DFEAT;
    const long long nE = in_sizes[2] / 2;

    // d_out doubles as the projection buffer (fully overwritten every call).
    float* out_op = (float*)d_out;                       // [n_op, 128]
    float* out_m  = out_op + (size_t)n_op * DFEAT;       // [n_m, 128]

    // Workspace: sums + degree counters (zeroed each call).
    float* sum_seq = (float*)d_ws;                       // [n_op, 128]
    float* sum_op2 = sum_seq + (size_t)n_op * DFEAT;     // [n_op, 128]
    float* sum_m   = sum_op2 + (size_t)n_op * DFEAT;     // [n_m, 128]
    float* deg_seq = sum_m   + (size_t)n_m  * DFEAT;     // [n_op]
    float* deg_op2 = deg_seq + n_op;                     // [n_op]
    float* deg_m   = deg_op2 + n_op;                     // [n_m]
    size_t zero_floats = 2 * (size_t)n_op * DFEAT + (size_t)n_m * DFEAT
                       + 2 * (size_t)n_op + (size_t)n_m;
    hipMemsetAsync(d_ws, 0, zero_floats * sizeof(float), stream);

    // Projections (WMMA GEMMs): 4 waves/block, 64 rows/block.
    gemm128_bias_wmma<<<(n_op + 63) / 64, 128, 0, stream>>>(H_op, W_op, b_op, out_op, n_op);
    gemm128_bias_wmma<<<(n_m  + 63) / 64, 128, 0, stream>>>(H_m,  W_m,  b_m,  out_m,  n_m);

    // Mean aggregations (scatter-add + degree).
    long long tot = nE * DFEAT;
    int sblocks = (int)((tot + 255) / 256);
    // op->op sequence edges: msgs = out_op[src], scatter to dst (op nodes)
    scatter_accum<<<sblocks, 256, 0, stream>>>(out_op, E_seq, E_seq + nE,
                                               sum_seq, deg_seq, nE);
    // op->machine: msgs = out_op[src_op], scatter to dst_m (machine nodes)
    scatter_accum<<<sblocks, 256, 0, stream>>>(out_op, E_op2m, E_op2m + nE,
                                               sum_m, deg_m, nE);
    // machine->op: msgs = out_m[dst_m], scatter to src_op (op nodes)
    scatter_accum<<<sblocks, 256, 0, stream>>>(out_m, E_op2m + nE, E_op2m,
                                               sum_op2, deg_op2, nE);

    // Fused add + ReLU (in place over the projections).
    long long nop_e = (long long)n_op * DFEAT;
    long long nm_e  = (long long)n_m * DFEAT;
    combine2_relu<<<(int)((nop_e + 255) / 256), 256, 0, stream>>>(
        out_op, sum_seq, deg_seq, sum_op2, deg_op2, nop_e);
    combine1_relu<<<(int)((nm_e + 255) / 256), 256, 0, stream>>>(
        out_m, sum_m, deg_m, nm_e);
}